// GNN_31473520345321
// MI455X (gfx1250) — compile-verified
//
#include <hip/hip_runtime.h>
#include <math.h>

// ---------------------------------------------------------------------------
// Quantum-GNN for MI455X (gfx1250, wave32).
//   * edge circuit (10 qubits, theta shared across batch) -> precompute the
//     1024x1024 unitary U once, then batched statevector evolution is a GEMM
//     done with V_WMMA_F32_16X16X4_F32, fused with the <Z_7> reduction.
//   * node circuit (15 qubits) -> full 128KB statevector per node in LDS
//     (CDNA5 WGP has 320KB), one workgroup of 1024 threads per node.
// ---------------------------------------------------------------------------

#define N_NODES 384
#define N_EDGES 1280

typedef __attribute__((ext_vector_type(2))) float v2f;
typedef __attribute__((ext_vector_type(8))) float v8f;

// ---- circuit op tables: T: 0=ry,1=cx ; A: theta idx / control ; B: wire / target
__constant__ int EOP_T[28] = {0,0,1,0,0,1,0,0,1,0,0,1,0,0,1,0,0,1,0,0,1,0,0,1,0,0,1,0};
__constant__ int EOP_A[28] = {0,1,0,2,3,3,4,5,5,6,7,6,8,9,8,10,11,1,12,13,9,14,15,2,16,17,4,18};
__constant__ int EOP_B[28] = {0,1,1,2,3,2,4,5,4,6,7,7,8,9,9,1,2,2,7,9,7,2,4,4,4,7,7,7};

__constant__ int NOP_T[50] = {0,0,1,0,0,1,0,0,1,0,0,1,0,0,1,0,0,1,0,0,1,0,0,0,1,
                              0,0,1,0,0,1,0,0,1,0,0,1,0,0,1,0,0,0,0,1,0,0,1,0,0};
__constant__ int NOP_A[50] = {0,1,0,2,3,3,4,5,4,6,7,7,8,9,8,10,11,11,12,13,8,14,15,16,1,
                              14,15,6,16,17,9,18,19,9,19,20,2,21,22,13,23,24,25,26,0,27,28,14,29,30};
__constant__ int NOP_B[50] = {0,1,1,2,3,2,4,5,5,6,7,6,8,9,9,10,11,10,12,13,9,14,1,2,2,
                              5,6,5,9,10,10,13,14,10,2,5,5,10,13,10,5,10,0,5,5,10,14,10,5,10};

__device__ __forceinline__ unsigned ins0(unsigned p, int b) {
    return ((p >> b) << (b + 1)) | (p & ((1u << b) - 1u));
}

// ---- H0 = [sigmoid(X@W)*2pi , X]  -> (384,5)
__global__ void k_init_H(const float* __restrict__ X, const float* __restrict__ W,
                         float* __restrict__ H) {
    int n = blockIdx.x * blockDim.x + threadIdx.x;
    if (n >= N_NODES) return;
    float x0 = X[n*3+0], x1 = X[n*3+1], x2 = X[n*3+2];
    float a0 = x0*W[0] + x1*W[2] + x2*W[4];
    float a1 = x0*W[1] + x1*W[3] + x2*W[5];
    const float TWO_PI = 6.28318530717958647692f;
    H[n*5+0] = TWO_PI / (1.0f + expf(-a0));
    H[n*5+1] = TWO_PI / (1.0f + expf(-a1));
    H[n*5+2] = x0; H[n*5+3] = x1; H[n*5+4] = x2;
}

// ---- recover snd/rcv indices from one-hot Ro/Ri (384 x 1280)
__global__ void k_extract_idx(const float* __restrict__ Ro, const float* __restrict__ Ri,
                              int* __restrict__ snd, int* __restrict__ rcv) {
    int e = blockIdx.x * blockDim.x + threadIdx.x;
    if (e >= N_EDGES) return;
    int s = 0, r = 0;
    for (int n = 0; n < N_NODES; ++n) {
        if (Ro[n*N_EDGES + e] > 0.5f) s = n;
        if (Ri[n*N_EDGES + e] > 0.5f) r = n;
    }
    snd[e] = s; rcv[e] = r;
}

// ---- build edge-circuit unitary: Ut[j][i] = U[i][j]; one column per block
__global__ void k_build_U(const float* __restrict__ theta, float* __restrict__ Ut) {
    __shared__ float st[1024];
    __shared__ float tc[19], ts[19];
    int col = blockIdx.x, tid = threadIdx.x;              // 512 threads
    if (tid < 19) { float h = theta[tid]*0.5f; tc[tid] = cosf(h); ts[tid] = sinf(h); }
    st[tid]       = (tid == col) ? 1.0f : 0.0f;
    st[tid + 512] = (tid + 512 == col) ? 1.0f : 0.0f;
    __syncthreads();
    for (int op = 0; op < 28; ++op) {
        if (EOP_T[op] == 0) {                             // RY, 512 pairs
            int wb = 9 - EOP_B[op];
            float c = tc[EOP_A[op]], s = ts[EOP_A[op]];
            unsigned i0 = ins0(tid, wb), i1 = i0 | (1u << wb);
            float a = st[i0], b = st[i1];
            st[i0] = c*a - s*b;
            st[i1] = s*a + c*b;
        } else if (tid < 256) {                           // CNOT: swap target for control=1
            int cb = 9 - EOP_A[op], tb = 9 - EOP_B[op];
            int lo = min(cb, tb), hi = max(cb, tb);
            unsigned i0 = ins0(ins0((unsigned)tid, lo), hi) | (1u << cb);
            unsigned i1 = i0 | (1u << tb);
            float a = st[i0], b = st[i1];
            st[i0] = b; st[i1] = a;
        }
        __syncthreads();
    }
    Ut[col*1024 + tid]       = st[tid];
    Ut[col*1024 + tid + 512] = st[tid + 512];
}

// ---- per-edge product state (1280 x 1024): one block per edge
__global__ void k_psi0(const float* __restrict__ H, const int* __restrict__ snd,
                       const int* __restrict__ rcv, float* __restrict__ Psi0) {
    __shared__ float pc[10], ps[10];
    int e = blockIdx.x, tid = threadIdx.x;                // 256 threads
    if (tid < 10) {
        int n = (tid < 5) ? snd[e] : rcv[e];
        int f = (tid < 5) ? tid : tid - 5;
        float h = H[n*5 + f] * 0.5f;
        pc[tid] = cosf(h); ps[tid] = sinf(h);
    }
    __syncthreads();
    for (int i = 0; i < 4; ++i) {
        int k = tid + i*256;
        float amp = 1.0f;
        #pragma unroll
        for (int w = 0; w < 10; ++w)
            amp *= ((k >> (9 - w)) & 1) ? ps[w] : pc[w];
        Psi0[e*1024 + k] = amp;
    }
}

__global__ void k_zero(float* __restrict__ p, int n) {
    int i = blockIdx.x * blockDim.x + threadIdx.x;
    if (i < n) p[i] = 0.0f;
}

// ---- WMMA GEMM: final = Psi0(1280x1024) @ Ut(1024x1024, row j = column j of U),
//      fused with z[e] += sum_i sign(i) * final[e,i]^2  (Z on wire 7 -> bit 2)
//      grid (80, 8), block = 1 wave (32). Each wave: 16-row tile x 8 column tiles.
__global__ void k_gemm_expz(const float* __restrict__ A, const float* __restrict__ B,
                            float* __restrict__ z) {
    int lane = threadIdx.x;                // 0..31
    int m0 = blockIdx.x * 16;              // edge-row tile base
    int m  = lane & 15;                    // row within tile (A) / col within tile (B)
    int kb = (lane >> 4) * 2;              // K sub-offset for second half-wave
    // column index within tile = lane&15; tiles start at multiples of 16,
    // so sign bit ((n>>2)&1) is per-lane constant:
    float sgn = ((m >> 2) & 1) ? -1.0f : 1.0f;
    float part[8];
    #pragma unroll
    for (int r = 0; r < 8; ++r) part[r] = 0.0f;

    for (int t = 0; t < 8; ++t) {
        int n0 = (blockIdx.y * 8 + t) * 16;
        v8f acc = {};
        for (int k = 0; k < 1024; k += 4) {
            v2f av, bv;
            av.x = A[(m0 + m)*1024 + k + kb];
            av.y = A[(m0 + m)*1024 + k + kb + 1];
            bv.x = B[(k + kb)    *1024 + n0 + m];
            bv.y = B[(k + kb + 1)*1024 + n0 + m];
            acc = __builtin_amdgcn_wmma_f32_16x16x4_f32(
                false, av, false, bv, (short)0, acc, false, false);
        }
        #pragma unroll
        for (int r = 0; r < 8; ++r) part[r] += acc[r]*acc[r];
    }
    // reduce over the 16 columns held by each half-wave (xor masks 1..8 stay in-half)
    #pragma unroll
    for (int r = 0; r < 8; ++r) {
        float v = sgn * part[r];
        for (int off = 1; off < 16; off <<= 1) v += __shfl_xor(v, off, 32);
        if (lane == 0)       atomicAdd(&z[m0 + r],     v);   // M = r
        else if (lane == 16) atomicAdd(&z[m0 + 8 + r], v);   // M = r + 8
    }
}

__global__ void k_finish_e(const float* __restrict__ z, float* __restrict__ e) {
    int i = blockIdx.x * blockDim.x + threadIdx.x;
    if (i < N_EDGES) e[i] = 0.5f * (1.0f - z[i]);
}

// ---- scatter-add messages: mi[rcv] += e*H[snd], mo[snd] += e*H[rcv]
__global__ void k_scatter(const float* __restrict__ H, const float* __restrict__ eW,
                          const int* __restrict__ snd, const int* __restrict__ rcv,
                          float* __restrict__ mi, float* __restrict__ mo) {
    int e = blockIdx.x * blockDim.x + threadIdx.x;
    if (e >= N_EDGES) return;
    int s = snd[e], r = rcv[e];
    float w = eW[e];
    #pragma unroll
    for (int f = 0; f < 5; ++f) {
        atomicAdd(&mi[r*5 + f], w * H[s*5 + f]);
        atomicAdd(&mo[s*5 + f], w * H[r*5 + f]);
    }
}

// ---- node circuit: 15 qubits, 32768-amp state fully in LDS (128KB of 320KB/WGP)
//      one block of 1024 threads per node; writes updated H row.
__global__ void __launch_bounds__(1024)
k_node_circ(const float* __restrict__ mi, const float* __restrict__ mo,
            const float* __restrict__ X, const float* __restrict__ theta,
            float* __restrict__ H) {
    extern __shared__ float st[];          // 32768 floats
    __shared__ float tc[31], ts[31];
    __shared__ float pc[15], ps[15];
    __shared__ float zred[2];
    int node = blockIdx.x, tid = threadIdx.x;
    if (tid < 31) { float h = theta[tid]*0.5f; tc[tid] = cosf(h); ts[tid] = sinf(h); }
    if (tid < 15) {
        float ang;
        if (tid < 5)       ang = mi[node*5 + tid];
        else if (tid < 10) ang = mo[node*5 + tid - 5];
        else               ang = H [node*5 + tid - 10];
        float h = ang * 0.5f;
        pc[tid] = cosf(h); ps[tid] = sinf(h);
    }
    if (tid == 0) { zred[0] = 0.0f; zred[1] = 0.0f; }
    __syncthreads();
    // product state
    for (int i = 0; i < 32; ++i) {
        int k = tid + i*1024;
        float amp = 1.0f;
        #pragma unroll
        for (int w = 0; w < 15; ++w)
            amp *= ((k >> (14 - w)) & 1) ? ps[w] : pc[w];
        st[k] = amp;
    }
    __syncthreads();
    for (int op = 0; op < 50; ++op) {
        if (NOP_T[op] == 0) {                  // RY: 16384 pairs, 16 per thread
            int wb = 14 - NOP_B[op];
            float c = tc[NOP_A[op]], s = ts[NOP_A[op]];
            for (int i = 0; i < 16; ++i) {
                unsigned p = tid + i*1024;
                unsigned i0 = ins0(p, wb), i1 = i0 | (1u << wb);
                float a = st[i0], b = st[i1];
                st[i0] = c*a - s*b;
                st[i1] = s*a + c*b;
            }
        } else {                               // CNOT: 8192 swap sites, 8 per thread
            int cb = 14 - NOP_A[op], tb = 14 - NOP_B[op];
            int lo = min(cb, tb), hi = max(cb, tb);
            for (int i = 0; i < 8; ++i) {
                unsigned p = tid + i*1024;
                unsigned i0 = ins0(ins0(p, lo), hi) | (1u << cb);
                unsigned i1 = i0 | (1u << tb);
                float a = st[i0], b = st[i1];
                st[i0] = b; st[i1] = a;
            }
        }
        __syncthreads();
    }
    // <Z> on wires 5 (bit 9) and 10 (bit 4)
    float p5 = 0.0f, p10 = 0.0f;
    for (int i = 0; i < 32; ++i) {
        int k = tid + i*1024;
        float a = st[k], a2 = a*a;
        p5  += ((k >> 9) & 1) ? -a2 : a2;
        p10 += ((k >> 4) & 1) ? -a2 : a2;
    }
    atomicAdd(&zred[0], p5);
    atomicAdd(&zred[1], p10);
    __syncthreads();
    if (tid == 0) {
        const float PI_ = 3.14159265358979323846f;
        H[node*5+0] = PI_ * (1.0f - zred[0]);
        H[node*5+1] = PI_ * (1.0f - zred[1]);
        H[node*5+2] = X[node*3+0];
        H[node*5+3] = X[node*3+1];
        H[node*5+4] = X[node*3+2];
    }
}

extern "C" void kernel_launch(void* const* d_in, const int* in_sizes, int n_in,
                              void* d_out, int out_size, void* d_ws, size_t ws_size,
                              hipStream_t stream) {
    const float* X    = (const float*)d_in[0];   // (384,3)
    const float* Ri   = (const float*)d_in[1];   // (384,1280)
    const float* Ro   = (const float*)d_in[2];   // (384,1280)
    const float* W    = (const float*)d_in[3];   // (3,2)
    const float* th_e = (const float*)d_in[4];   // (19,)
    const float* th_n = (const float*)d_in[5];   // (31,)
    float* out = (float*)d_out;                  // (1280,)

    char* ws = (char*)d_ws;
    float* H    = (float*)(ws + 0);              // 384*5
    int*   snd  = (int*)  (ws + 8192);           // 1280
    int*   rcv  = (int*)  (ws + 16384);          // 1280
    float* mi   = (float*)(ws + 24576);          // 384*5
    float* mo   = (float*)(ws + 32768);          // 384*5
    float* z    = (float*)(ws + 40960);          // 1280
    float* eW   = (float*)(ws + 49152);          // 1280
    float* Ut   = (float*)(ws + 65536);          // 1024*1024  (4 MB)
    float* Psi0 = (float*)(ws + 65536 + 4194304);// 1280*1024  (5 MB)

    k_init_H<<<3, 128, 0, stream>>>(X, W, H);
    k_extract_idx<<<10, 128, 0, stream>>>(Ro, Ri, snd, rcv);
    k_build_U<<<1024, 512, 0, stream>>>(th_e, Ut);   // theta_edge fixed: built once

    dim3 gg(80, 8);
    // pass 0
    k_psi0<<<N_EDGES, 256, 0, stream>>>(H, snd, rcv, Psi0);
    k_zero<<<5, 256, 0, stream>>>(z, N_EDGES);
    k_gemm_expz<<<gg, 32, 0, stream>>>(Psi0, Ut, z);
    k_finish_e<<<5, 256, 0, stream>>>(z, eW);

    for (int it = 0; it < 2; ++it) {
        k_zero<<<16, 256, 0, stream>>>(mi, 4096);    // covers mi..mo region
        k_scatter<<<10, 128, 0, stream>>>(H, eW, snd, rcv, mi, mo);
        k_node_circ<<<N_NODES, 1024, 32768 * sizeof(float), stream>>>(mi, mo, X, th_n, H);
        k_psi0<<<N_EDGES, 256, 0, stream>>>(H, snd, rcv, Psi0);
        k_zero<<<5, 256, 0, stream>>>(z, N_EDGES);
        k_gemm_expz<<<gg, 32, 0, stream>>>(Psi0, Ut, z);
        k_finish_e<<<5, 256, 0, stream>>>(z, (it == 1) ? out : eW);
    }
}